// MoEGate_71176198029864
// MI455X (gfx1250) — compile-verified
//
#include <hip/hip_runtime.h>
#include <hip/hip_bf16.h>

typedef float v2f __attribute__((ext_vector_type(2)));
typedef float v8f __attribute__((ext_vector_type(8)));

#define NTOK   16384
#define DIM    2048
#define NEXP   64
#define TOPK   8
#define ALPHA  0.001f

// ---------------------------------------------------------------------------
// Zero the 128-float workspace (64 score sums + 64 counts) each call.
// ---------------------------------------------------------------------------
__global__ void moe_gate_zero(float* ws) {
    ws[threadIdx.x] = 0.0f;
}

// ---------------------------------------------------------------------------
// Main kernel: fp32 WMMA router GEMM + softmax + top-8 + aux-loss partials.
// Block = 256 threads (8 waves). Each wave owns a 16-token tile and computes
// all 64 expert logits with v_wmma_f32_16x16x4_f32.
// Grid = 16384 / (8*16) = 128 blocks.
// ---------------------------------------------------------------------------
__global__ __launch_bounds__(256) void moe_gate_main(
    const float* __restrict__ hidden,   // (16384, 2048)
    const float* __restrict__ weight,   // (64, 2048)
    float* __restrict__ outW,           // (16384, 8)
    int*   __restrict__ outI,           // (16384, 8)
    float* __restrict__ gScore,         // (64) global prob sums
    float* __restrict__ gCnt)           // (64) global counts (as float)
{
    __shared__ float ldsL[8 * 16 * NEXP];   // 128 tokens x 64 experts logits
    __shared__ float ldsScore[NEXP];
    __shared__ int   ldsCnt[NEXP];

    const int tid = threadIdx.x;
    if (tid < NEXP) { ldsScore[tid] = 0.0f; ldsCnt[tid] = 0; }

    const int wave = tid >> 5;
    const int lane = tid & 31;
    const int lo   = lane & 15;   // row (A) / column (B) within 16
    const int hi   = lane >> 4;   // selects K pair {0,1} vs {2,3}
    const int tokenBase = (blockIdx.x * 8 + wave) * 16;

    // A fragment: lane (lo,hi) holds A[lo][2*hi + {0,1}] per ISA 16x4 fp32 layout
    const float* aPtr  = hidden + (size_t)(tokenBase + lo) * DIM + 2 * hi;
    // B fragments (K x 16 experts), one per 16-expert tile
    const float* bPtr0 = weight + (size_t)lo * DIM + 2 * hi;
    const float* bPtr1 = bPtr0 + (size_t)16 * DIM;
    const float* bPtr2 = bPtr0 + (size_t)32 * DIM;
    const float* bPtr3 = bPtr0 + (size_t)48 * DIM;

    v8f acc0 = {}, acc1 = {}, acc2 = {}, acc3 = {};

    for (int k0 = 0; k0 < DIM; k0 += 32) {
        // stream A ahead (global_prefetch_b8); W is L2-resident (512 KB)
        __builtin_prefetch(aPtr + k0 + 1024, 0, 0);
#pragma unroll
        for (int s = 0; s < 32; s += 4) {
            const int k = k0 + s;
            v2f a  = *(const v2f*)(aPtr  + k);
            v2f b0 = *(const v2f*)(bPtr0 + k);
            v2f b1 = *(const v2f*)(bPtr1 + k);
            v2f b2 = *(const v2f*)(bPtr2 + k);
            v2f b3 = *(const v2f*)(bPtr3 + k);
            acc0 = __builtin_amdgcn_wmma_f32_16x16x4_f32(false, a, false, b0,
                       (short)0, acc0, false, false);
            // same instruction repeats with identical A -> reuse_a hint legal
            acc1 = __builtin_amdgcn_wmma_f32_16x16x4_f32(false, a, false, b1,
                       (short)0, acc1, true, false);
            acc2 = __builtin_amdgcn_wmma_f32_16x16x4_f32(false, a, false, b2,
                       (short)0, acc2, true, false);
            acc3 = __builtin_amdgcn_wmma_f32_16x16x4_f32(false, a, false, b3,
                       (short)0, acc3, true, false);
        }
    }

    // Scatter C fragments (VGPR j: lanes0-15 -> M=j, lanes16-31 -> M=j+8)
    // into LDS as [token][expert].
    float* tileBase = &ldsL[wave * 16 * NEXP];
#pragma unroll
    for (int j = 0; j < 8; ++j) {
        const int t = j + 8 * hi;
        tileBase[t * NEXP +  0 + lo] = acc0[j];
        tileBase[t * NEXP + 16 + lo] = acc1[j];
        tileBase[t * NEXP + 32 + lo] = acc2[j];
        tileBase[t * NEXP + 48 + lo] = acc3[j];
    }
    __syncthreads();

    // -------- per-token softmax + top-8 (one thread per token, 128 tokens) ---
    if (tid < 128) {
        const float* row = &ldsL[tid * NEXP];
        float lv[NEXP];
#pragma unroll
        for (int e = 0; e < NEXP; ++e) lv[e] = row[e];

        float m = lv[0];
#pragma unroll
        for (int e = 1; e < NEXP; ++e) m = fmaxf(m, lv[e]);

        float Z = 0.0f;
#pragma unroll
        for (int e = 0; e < NEXP; ++e) Z += __expf(lv[e] - m);
        const float invZ = 1.0f / Z;

        // per-expert softmax prob sums (rotate index: conflict-free across lanes)
#pragma unroll
        for (int e = 0; e < NEXP; ++e) {
            const int ei = (e + tid) & (NEXP - 1);
            atomicAdd(&ldsScore[ei], __expf(row[ei] - m) * invZ);
        }

        // top-8 by repeated argmax; strict '>' keeps lowest index on ties,
        // values emerge in descending order (matches lax.top_k).
        unsigned long long used = 0ull;
        float wv[TOPK]; int wi[TOPK]; float wsum = 0.0f;
#pragma unroll
        for (int p = 0; p < TOPK; ++p) {
            float best = -3.402823466e38f; int bidx = 0;
#pragma unroll
            for (int e = 0; e < NEXP; ++e) {
                const bool free_e = ((used >> e) & 1ull) == 0ull;
                if (free_e && lv[e] > best) { best = lv[e]; bidx = e; }
            }
            used |= (1ull << bidx);
            const float w = __expf(best - m);   // renorm == softmax over top-8
            wv[p] = w; wi[p] = bidx; wsum += w;
            atomicAdd(&ldsCnt[bidx], 1);
        }
        const float inv = 1.0f / wsum;
        const int g = blockIdx.x * 128 + tid;
#pragma unroll
        for (int p = 0; p < TOPK; ++p) {
            outW[g * TOPK + p] = wv[p] * inv;
            outI[g * TOPK + p] = wi[p];
        }
    }
    __syncthreads();

    // -------- flush block partials: 64 global atomics each ------------------
    if (tid < NEXP) {
        atomicAdd(&gScore[tid], ldsScore[tid]);
        atomicAdd(&gCnt[tid],   (float)ldsCnt[tid]);
    }
}

// ---------------------------------------------------------------------------
// Finalize aux loss: ALPHA * E * sum_e (cnt[e]/(N*K)) * (scoreSum[e]/N)
// ---------------------------------------------------------------------------
__global__ void moe_gate_final(const float* __restrict__ gScore,
                               const float* __restrict__ gCnt,
                               float* __restrict__ aux)
{
    __shared__ float red[NEXP];
    const int t = threadIdx.x;
    red[t] = (gCnt[t] * (1.0f / ((float)NTOK * (float)TOPK))) *
             (gScore[t] * (1.0f / (float)NTOK));
    __syncthreads();
    if (t == 0) {
        float s = 0.0f;
        for (int i = 0; i < NEXP; ++i) s += red[i];
        aux[0] = ALPHA * (float)NEXP * s;
    }
}

// ---------------------------------------------------------------------------
extern "C" void kernel_launch(void* const* d_in, const int* in_sizes, int n_in,
                              void* d_out, int out_size, void* d_ws, size_t ws_size,
                              hipStream_t stream) {
    const float* hidden = (const float*)d_in[0];   // (16384, 2048) f32
    const float* weight = (const float*)d_in[1];   // (64, 2048) f32

    float* outW = (float*)d_out;                       // 16384*8 weights
    int*   outI = (int*)d_out + NTOK * TOPK;           // 16384*8 indices (int32)
    float* aux  = (float*)d_out + 2 * NTOK * TOPK;     // scalar aux loss

    float* wsf = (float*)d_ws;          // [0..63] prob sums, [64..127] counts

    moe_gate_zero <<<1, 128, 0, stream>>>(wsf);
    moe_gate_main <<<NTOK / 128, 256, 0, stream>>>(hidden, weight, outW, outI,
                                                   wsf, wsf + NEXP);
    moe_gate_final<<<1, NEXP, 0, stream>>>(wsf, wsf + NEXP, aux);
}